// SGEmbedder_4398046511361
// MI455X (gfx1250) — compile-verified
//
#include <hip/hip_runtime.h>
#include <hip/hip_bf16.h>
#include <math.h>

typedef __attribute__((ext_vector_type(2))) float v2f;
typedef __attribute__((ext_vector_type(8))) float v8f;
typedef __attribute__((ext_vector_type(4))) unsigned int v4u;
typedef __attribute__((ext_vector_type(8))) int v8i;
typedef __attribute__((ext_vector_type(4))) int v4i;

#define DIM 128
#define LDA 132  // A tile LDS stride: 132 mod 64 = 4 -> conflict-free strided A reads
#define LDW 136  // W LDS stride: rows k and k+2 land on disjoint 16-bank sets for B reads

#if __has_builtin(__builtin_amdgcn_tensor_load_to_lds) && __has_builtin(__builtin_amdgcn_s_wait_tensorcnt)
#define USE_TDM 1
#else
#define USE_TDM 0
#endif

// ---------------- degree / normalization ----------------

__global__ void deg_init_kernel(float* __restrict__ deg, int N) {
  int i = blockIdx.x * blockDim.x + threadIdx.x;
  if (i < N) deg[i] = 1.0f;  // self loop contributes 1 to every node's degree
}

__global__ void deg_edge_kernel(float* __restrict__ deg,
                                const long long* __restrict__ dst, int E) {
  int e = blockIdx.x * blockDim.x + threadIdx.x;
  if (e < E) atomicAdd(&deg[(int)dst[e]], 1.0f);
}

__global__ void dinv_kernel(float* __restrict__ deg, int N) {
  int i = blockIdx.x * blockDim.x + threadIdx.x;
  if (i < N) deg[i] = rsqrtf(deg[i]);  // deg >= 1 always (self loops)
}

__global__ void norm_kernel(const float* __restrict__ dinv,
                            const long long* __restrict__ src,
                            const long long* __restrict__ dst,
                            float* __restrict__ norm, int E) {
  int e = blockIdx.x * blockDim.x + threadIdx.x;
  if (e < E) norm[e] = dinv[(int)src[e]] * dinv[(int)dst[e]];
}

// ---------------- S @ h propagation (float4 = b128 per lane) ----------------

// self-loop term: out[i] = dinv[i]^2 * in[i]  (also initializes `out`)
__global__ void prop_self_kernel(const float4* __restrict__ in4,
                                 float4* __restrict__ out4,
                                 const float* __restrict__ dinv, int N) {
  size_t i = (size_t)blockIdx.x * blockDim.x + threadIdx.x;  // over N*32
  if (i < (size_t)N * 32) {
    float w = dinv[i >> 5];
    w *= w;
    float4 v = in4[i];
    out4[i] = make_float4(w * v.x, w * v.y, w * v.z, w * v.w);
  }
}

// edge term: out[dst] += norm[e] * in[src]; one thread per (edge, 4 dims)
__global__ void prop_edge_kernel(const float4* __restrict__ in4,
                                 float* __restrict__ out,
                                 const float* __restrict__ norm,
                                 const long long* __restrict__ src,
                                 const long long* __restrict__ dst, int E) {
  size_t i = (size_t)blockIdx.x * blockDim.x + threadIdx.x;  // over E*32
  if (i >= (size_t)E * 32) return;
  int e = (int)(i >> 5);
  int q = (int)(i & 31);
  int s = (int)src[e];
  int t = (int)dst[e];
  float w = norm[e];
  float4 v = in4[(size_t)s * 32 + q];
  float* o = out + (size_t)t * DIM + q * 4;
  atomicAdd(o + 0, w * v.x);
  atomicAdd(o + 1, w * v.y);
  atomicAdd(o + 2, w * v.z);
  atomicAdd(o + 3, w * v.w);
}

// ---------------- TDM: async 2D tile load Global -> LDS ----------------
// Builds a D# (groups 0/1; remaining groups zero => 2D tensor, count=1) and
// issues TENSOR_LOAD_TO_LDS. LDS row padding done by the TDM pad engine.
#if USE_TDM
__device__ __forceinline__ void tdm_load_2d(unsigned lds_off, const void* gaddr,
                                            unsigned dim0, unsigned dim1,
                                            unsigned tile0, unsigned tile1,
                                            unsigned stride0,
                                            unsigned pad_interval_code,
                                            unsigned pad_amount_code) {
  unsigned long long ga = (unsigned long long)(size_t)gaddr;
  v4u g0;
  g0[0] = 1u;                                   // count=1, user descriptor
  g0[1] = lds_off;                              // lds_addr (bytes)
  g0[2] = (unsigned)(ga & 0xFFFFFFFFu);         // global_addr[31:0]
  g0[3] = (unsigned)((ga >> 32) & 0x01FFFFFFu)  // global_addr[56:32]
          | (2u << 30);                         // type = 2 ("image")
  v8i g1;
  g1[0] = (int)((2u << 16)                      // data_size = 4 bytes
                | (1u << 20)                    // pad_enable
                | (pad_interval_code << 22)     // DWORDs between pads
                | (pad_amount_code << 25));     // pad DWORDs - 1
  g1[1] = (int)((dim0 & 0xFFFFu) << 16);        // [63:48] = tensor_dim0 lo16
  g1[2] = (int)(((dim0 >> 16) & 0xFFFFu) | ((dim1 & 0xFFFFu) << 16));
  g1[3] = (int)(((dim1 >> 16) & 0xFFFFu) | ((tile0 & 0xFFFFu) << 16));
  g1[4] = (int)(tile1 & 0xFFFFu);               // tile_dim1 (tile_dim2 = 0)
  g1[5] = (int)stride0;                         // tensor_dim0_stride lo32
  g1[6] = 0;                                    // stride0 hi16 | stride1 lo16
  g1[7] = 0;
  v4i z4 = {0, 0, 0, 0};
  v8i z8 = {0, 0, 0, 0, 0, 0, 0, 0};
  // 6-arg form (clang-23 / therock-10.0 headers): extra zero group before cpol
  __builtin_amdgcn_tensor_load_to_lds(g0, g1, z4, z4, z8, 0);
}
#endif

// ---------------- Y = tanh(H @ W + b) via V_WMMA_F32_16X16X4_F32 ----------------
// Block = 256 threads = 8 waves covering 16 rows x 128 cols; wave w owns the
// 16x16 tile at columns [16w, 16w+16). TDM stages the 16x128 H tile (pad->
// stride 132) and the full 128x128 W (pad->stride 136) into LDS; k-loop is
// then pure DS reads + chained WMMA.
__global__ __launch_bounds__(256)
void gemm_bias_tanh_wmma(const float* __restrict__ H, const float* __restrict__ W,
                         const float* __restrict__ bias, float* __restrict__ Y,
                         int N) {
  __shared__ float sA[16 * LDA];
  __shared__ float sW[DIM * LDW];

  const int tid = threadIdx.x;
  const size_t rowBase = (size_t)blockIdx.x * 16;
  const unsigned rows =
      (rowBase + 16 <= (size_t)N) ? 16u : (unsigned)((size_t)N - rowBase);

#if USE_TDM
  if (tid < 32) {  // one wave issues the DMA; waits on its TENSORcnt
    // H tile: 16 rows x 128 cols, pad 4 DWORDs every 128 -> LDS stride 132
    tdm_load_2d((unsigned)(size_t)&sA[0], H + rowBase * DIM,
                DIM, rows, DIM, rows, DIM, /*interval 128dw*/ 6, /*4dw*/ 3);
    // W: 128 x 128, pad 8 DWORDs every 128 -> LDS stride 136
    tdm_load_2d((unsigned)(size_t)&sW[0], W,
                DIM, DIM, DIM, DIM, DIM, /*interval 128dw*/ 6, /*8dw*/ 7);
    __builtin_amdgcn_s_wait_tensorcnt(0);
  }
  __syncthreads();
#else
  for (int i = tid; i < 16 * DIM; i += 256) {
    int r = i >> 7, c = i & (DIM - 1);
    size_t row = rowBase + r;
    if (row >= (size_t)N) row = (size_t)N - 1;
    sA[r * LDA + c] = H[row * DIM + c];
  }
  for (int i = tid; i < DIM * DIM; i += 256) {
    int r = i >> 7, c = i & (DIM - 1);
    sW[r * LDW + c] = W[i];
  }
  __syncthreads();
#endif

  const int wave = tid >> 5;
  const int lane = tid & 31;
  const int m  = lane & 15;        // M (for A) / N (for B,C,D)
  const int kh = (lane >> 4) << 1; // 0 or 2: which K pair this half-wave holds
  const int colBase = wave << 4;

  v8f c = {};
#pragma unroll
  for (int k = 0; k < DIM; k += 4) {
    v2f a, b;
    a.x = sA[m * LDA + k + kh];
    a.y = sA[m * LDA + k + kh + 1];
    b.x = sW[(k + kh) * LDW + colBase + m];
    b.y = sW[(k + kh + 1) * LDW + colBase + m];
    c = __builtin_amdgcn_wmma_f32_16x16x4_f32(
        /*neg_a=*/false, a, /*neg_b=*/false, b,
        /*c_mod=*/(short)0, c, /*reuse_a=*/false, /*reuse_b=*/false);
  }

  const float bv = bias[colBase + m];
  const int rowOff = (lane >> 4) << 3;  // lanes 16-31 hold rows +8
#pragma unroll
  for (int r = 0; r < 8; r++) {
    size_t row = rowBase + rowOff + r;
    if (row < (size_t)N)
      Y[row * DIM + colBase + m] = tanhf(c[r] + bv);
  }
}

// ---------------- global mean pool ----------------

__global__ void pool_zero_kernel(float* __restrict__ out, float* __restrict__ cnt,
                                 int G) {
  int i = blockIdx.x * blockDim.x + threadIdx.x;
  if (i < G * DIM) out[i] = 0.0f;
  if (i < G) cnt[i] = 0.0f;
}

__global__ void pool_scatter_kernel(const float4* __restrict__ h4,
                                    const long long* __restrict__ batch,
                                    float* __restrict__ out,
                                    float* __restrict__ cnt, int N) {
  size_t i = (size_t)blockIdx.x * blockDim.x + threadIdx.x;  // over N*32
  if (i >= (size_t)N * 32) return;
  int node = (int)(i >> 5);
  int q = (int)(i & 31);
  int g = (int)batch[node];
  float4 v = h4[i];
  float* o = out + (size_t)g * DIM + q * 4;
  atomicAdd(o + 0, v.x);
  atomicAdd(o + 1, v.y);
  atomicAdd(o + 2, v.z);
  atomicAdd(o + 3, v.w);
  if (q == 0) atomicAdd(&cnt[g], 1.0f);
}

__global__ void pool_div_kernel(float* __restrict__ out,
                                const float* __restrict__ cnt, int G) {
  int i = blockIdx.x * blockDim.x + threadIdx.x;
  if (i < G * DIM) out[i] /= fmaxf(cnt[i >> 7], 1.0f);
}

// ---------------- driver ----------------

extern "C" void kernel_launch(void* const* d_in, const int* in_sizes, int n_in,
                              void* d_out, int out_size, void* d_ws, size_t ws_size,
                              hipStream_t stream) {
  const float*     x     = (const float*)d_in[0];
  const long long* ei    = (const long long*)d_in[1];
  const long long* batch = (const long long*)d_in[2];
  const float*     W1    = (const float*)d_in[3];
  const float*     b1    = (const float*)d_in[4];
  const float*     W2    = (const float*)d_in[5];
  const float*     b2    = (const float*)d_in[6];

  const int N = in_sizes[0] / DIM;   // 500,000
  const int E = in_sizes[1] / 2;     // 2,000,000
  const int G = out_size / DIM;      // 1,024
  const long long* src = ei;
  const long long* dst = ei + E;

  // workspace layout
  char* ws = (char*)d_ws;
  const size_t hbytes = (size_t)N * DIM * sizeof(float);  // 256 MB
  float* hA   = (float*)ws;
  float* hB   = (float*)(ws + hbytes);
  float* dinv = (float*)(ws + 2 * hbytes);  // deg -> dinv in place
  float* norm = dinv + N;
  float* cnt  = norm + E;

  const int T = 256;
  const unsigned gN   = (unsigned)((N + T - 1) / T);
  const unsigned gE   = (unsigned)((E + T - 1) / T);
  const size_t n4 = (size_t)N * 32;
  const size_t e4 = (size_t)E * 32;
  const unsigned gN4  = (unsigned)((n4 + T - 1) / T);
  const unsigned gE4  = (unsigned)((e4 + T - 1) / T);
  const unsigned gGEMM = (unsigned)((N + 15) / 16);
  const unsigned gGD  = (unsigned)((G * DIM + T - 1) / T);

  // D^{-1/2} and per-edge norm
  deg_init_kernel<<<gN, T, 0, stream>>>(dinv, N);
  deg_edge_kernel<<<gE, T, 0, stream>>>(dinv, dst, E);
  dinv_kernel<<<gN, T, 0, stream>>>(dinv, N);
  norm_kernel<<<gE, T, 0, stream>>>(dinv, src, dst, norm, E);

  // layer 1: S^2 x -> tanh(. @ W1 + b1)
  prop_self_kernel<<<gN4, T, 0, stream>>>((const float4*)x, (float4*)hA, dinv, N);
  prop_edge_kernel<<<gE4, T, 0, stream>>>((const float4*)x, hA, norm, src, dst, E);
  prop_self_kernel<<<gN4, T, 0, stream>>>((const float4*)hA, (float4*)hB, dinv, N);
  prop_edge_kernel<<<gE4, T, 0, stream>>>((const float4*)hA, hB, norm, src, dst, E);
  gemm_bias_tanh_wmma<<<gGEMM, 256, 0, stream>>>(hB, W1, b1, hA, N);

  // layer 2: S^2 h -> tanh(. @ W2 + b2)
  prop_self_kernel<<<gN4, T, 0, stream>>>((const float4*)hA, (float4*)hB, dinv, N);
  prop_edge_kernel<<<gE4, T, 0, stream>>>((const float4*)hA, hB, norm, src, dst, E);
  prop_self_kernel<<<gN4, T, 0, stream>>>((const float4*)hB, (float4*)hA, dinv, N);
  prop_edge_kernel<<<gE4, T, 0, stream>>>((const float4*)hB, hA, norm, src, dst, E);
  gemm_bias_tanh_wmma<<<gGEMM, 256, 0, stream>>>(hA, W2, b2, hB, N);

  // global mean pool into d_out
  float* out = (float*)d_out;
  pool_zero_kernel<<<gGD, T, 0, stream>>>(out, cnt, G);
  pool_scatter_kernel<<<gN4, T, 0, stream>>>((const float4*)hB, batch, out, cnt, N);
  pool_div_kernel<<<gGD, T, 0, stream>>>(out, cnt, G);
}